// ASTGNNFactorModel_38010460569740
// MI455X (gfx1250) — compile-verified
//
#include <hip/hip_runtime.h>

// ASTGNN TGC stack on gfx1250, fp32 WMMA (V_WMMA_F32_16X16X4_F32), flash-softmax.
// B=4, N=4000, K=16, F_IN=32, F_HID=64, F_OUT=32.

typedef __attribute__((ext_vector_type(2))) float v2f;
typedef __attribute__((ext_vector_type(8))) float v8f;

#define B_    4
#define N_    4000
#define K_    16
#define FIN_  32
#define FHID_ 64
#define FOUT_ 32
#define TILES_ (N_ / 16)          // 250 row-tiles per batch
#define GTILES_ (B_ * TILES_)     // 1000 waves of work

static __device__ __forceinline__ v8f wmma4(v2f a, v2f b, v8f c) {
  // D(16x16,f32) = A(16x4,f32) * B(4x16,f32) + C
  return __builtin_amdgcn_wmma_f32_16x16x4_f32(
      /*neg_a=*/false, a, /*neg_b=*/false, b,
      /*c_mod=*/(short)0, c, /*reuse_a=*/false, /*reuse_b=*/false);
}

// Reduce across the 16 lanes of each half-wave (xor masks < 16 stay in-half).
// Matches the WMMA C/D layout: after this, every lane holds the row statistic
// for row (e + 8*half).
static __device__ __forceinline__ void rowmax16(float v[8]) {
#pragma unroll
  for (int m = 1; m <= 8; m <<= 1)
#pragma unroll
    for (int e = 0; e < 8; ++e) v[e] = fmaxf(v[e], __shfl_xor(v[e], m, 32));
}
static __device__ __forceinline__ void rowsum16(float v[8]) {
#pragma unroll
  for (int m = 1; m <= 8; m <<= 1)
#pragma unroll
    for (int e = 0; e < 8; ++e) v[e] += __shfl_xor(v[e], m, 32);
}

// ---------------- Kernel 1: Y1 = X @ W1  [B,N,64] ----------------
__global__ __launch_bounds__(256) void k_y1(const float* __restrict__ X,
                                            const float* __restrict__ W1,
                                            float* __restrict__ Y1) {
  int idx = blockIdx.x * 256 + threadIdx.x;
  if (idx >= B_ * N_ * FHID_) return;
  int o = idx & (FHID_ - 1);
  int row = idx >> 6;  // b*N + n
  const float* x = X + (size_t)row * FIN_;
  float acc = 0.f;
#pragma unroll
  for (int i = 0; i < FIN_; ++i) acc = fmaf(x[i], W1[i * FHID_ + o], acc);
  Y1[idx] = acc;
}

// ---------------- Kernel 2: layer 1 (mode 'add') ----------------
// Per wave: 16 rows. Flash softmax over sim = softmax(relu(M M^T)),
// Num = sim @ Y1 accumulated online, Z1 = relu(Y1 + Num + b1),
// Y2 = Z1 @ W2 via WMMA. Stores Y2 and softmax stats (rmax, rsum).
__global__ __launch_bounds__(128) void k_layer1(
    const float* __restrict__ M, const float* __restrict__ Y1,
    const float* __restrict__ W2, const float* __restrict__ b1,
    float* __restrict__ Y2, float* __restrict__ rmaxg,
    float* __restrict__ rsumg) {
  __shared__ float pbuf[4][16 * 17];   // P tile bounce (D-layout -> A-layout)
  __shared__ float zbuf[4][16 * 65];   // Z1 tile bounce

  const int lane = threadIdx.x & 31;
  const int wave = threadIdx.x >> 5;
  const int half = lane >> 4;       // 0|1
  const int lm   = lane & 15;       // 0..15
  const int c0   = half * 2;        // K sub-offset for 32-bit A/B frags

  int gtile = blockIdx.x * 4 + wave;
  if (gtile >= GTILES_) return;
  const int b  = gtile / TILES_;
  const int r0 = (gtile % TILES_) * 16;
  const float* Mb = M + (size_t)b * N_ * K_;
  float* pb = pbuf[wave];
  float* zb = zbuf[wave];

  // A-frags for this tile's 16 rows of M (16x16 as 4 K-steps)
  v2f aM[4];
#pragma unroll
  for (int s = 0; s < 4; ++s) {
    const float* p = Mb + (size_t)(r0 + lm) * K_ + 4 * s + c0;
    aM[s].x = p[0];
    aM[s].y = p[1];
  }

  float rmax_r[8], rsum_r[8];
#pragma unroll
  for (int e = 0; e < 8; ++e) { rmax_r[e] = 0.f; rsum_r[e] = 0.f; }  // relu => max >= 0
  v8f num[4];
#pragma unroll
  for (int t = 0; t < 4; ++t) num[t] = 0.0f;

  for (int m0 = 0; m0 < N_; m0 += 16) {
    if (m0 + 16 < N_) __builtin_prefetch(Mb + (size_t)(m0 + 16) * K_, 0, 1);

    // logits tile L = relu(M_rows @ M_chunk^T), K=16 in 4 WMMA steps
    v2f bM[4];
#pragma unroll
    for (int s = 0; s < 4; ++s) {
      const float* p = Mb + (size_t)(m0 + lm) * K_ + 4 * s + c0;
      bM[s].x = p[0];
      bM[s].y = p[1];
    }
    v8f Lv = 0.0f;
#pragma unroll
    for (int s = 0; s < 4; ++s) Lv = wmma4(aM[s], bM[s], Lv);

    float l[8], cm[8];
#pragma unroll
    for (int e = 0; e < 8; ++e) { l[e] = fmaxf(Lv[e], 0.f); cm[e] = l[e]; }
    rowmax16(cm);

    // online softmax update
    float mnew[8], sc[8], p[8], ps[8];
#pragma unroll
    for (int e = 0; e < 8; ++e) {
      mnew[e] = fmaxf(rmax_r[e], cm[e]);
      sc[e]   = __expf(rmax_r[e] - mnew[e]);
      p[e]    = __expf(l[e] - mnew[e]);
      ps[e]   = p[e];
      rmax_r[e] = mnew[e];
    }
    rowsum16(ps);
#pragma unroll
    for (int e = 0; e < 8; ++e) rsum_r[e] = rsum_r[e] * sc[e] + ps[e];
#pragma unroll
    for (int t = 0; t < 4; ++t)
#pragma unroll
      for (int e = 0; e < 8; ++e) num[t][e] *= sc[e];

    // bounce P (D-layout) through LDS, reload as A-frags
#pragma unroll
    for (int e = 0; e < 8; ++e) pb[(e + 8 * half) * 17 + lm] = p[e];
    v2f pa[4];
#pragma unroll
    for (int s = 0; s < 4; ++s) {
      pa[s].x = pb[lm * 17 + 4 * s + c0];
      pa[s].y = pb[lm * 17 + 4 * s + c0 + 1];
    }

    // Num(16x64) += P(16x16) @ Y1_chunk(16x64)
#pragma unroll
    for (int t = 0; t < 4; ++t)
#pragma unroll
      for (int s = 0; s < 4; ++s) {
        const float* q =
            Y1 + ((size_t)(b * N_ + m0 + 4 * s + c0)) * FHID_ + 16 * t + lm;
        v2f by;
        by.x = q[0];
        by.y = q[FHID_];
        num[t] = wmma4(pa[s], by, num[t]);
      }
  }

  float inv[8];
#pragma unroll
  for (int e = 0; e < 8; ++e) inv[e] = 1.0f / rsum_r[e];

  // Z1 = relu(Y1 + sim@Y1 + b1) -> LDS
#pragma unroll
  for (int t = 0; t < 4; ++t)
#pragma unroll
    for (int e = 0; e < 8; ++e) {
      int row = r0 + e + 8 * half, col = 16 * t + lm;
      float z = Y1[((size_t)b * N_ + row) * FHID_ + col] + num[t][e] * inv[e] +
                b1[col];
      zb[(e + 8 * half) * 65 + col] = fmaxf(z, 0.f);
    }

  // Y2 = Z1 @ W2  (16x64 @ 64x32), K=64 in 16 WMMA steps per 16-col tile
  v8f acc2[2];
  acc2[0] = 0.0f;
  acc2[1] = 0.0f;
#pragma unroll
  for (int s = 0; s < 16; ++s) {
    v2f az;
    az.x = zb[lm * 65 + 4 * s + c0];
    az.y = zb[lm * 65 + 4 * s + c0 + 1];
#pragma unroll
    for (int t = 0; t < 2; ++t) {
      v2f bw;
      bw.x = W2[(4 * s + c0) * FOUT_ + 16 * t + lm];
      bw.y = W2[(4 * s + c0 + 1) * FOUT_ + 16 * t + lm];
      acc2[t] = wmma4(az, bw, acc2[t]);
    }
  }
#pragma unroll
  for (int t = 0; t < 2; ++t)
#pragma unroll
    for (int e = 0; e < 8; ++e)
      Y2[((size_t)b * N_ + r0 + e + 8 * half) * FOUT_ + 16 * t + lm] =
          acc2[t][e];

  if (lm == 0) {
#pragma unroll
    for (int e = 0; e < 8; ++e) {
      rmaxg[b * N_ + r0 + e + 8 * half] = rmax_r[e];
      rsumg[b * N_ + r0 + e + 8 * half] = rsum_r[e];
    }
  }
}

// ---------------- Kernel 3: layer 2 (mode 'subtract') ----------------
// out = Y2 - sim @ Y2 + b2, reusing stored softmax stats (single sweep).
__global__ __launch_bounds__(128) void k_layer2(
    const float* __restrict__ M, const float* __restrict__ Y2,
    const float* __restrict__ b2, const float* __restrict__ rmaxg,
    const float* __restrict__ rsumg, float* __restrict__ out) {
  __shared__ float pbuf[4][16 * 17];

  const int lane = threadIdx.x & 31;
  const int wave = threadIdx.x >> 5;
  const int half = lane >> 4;
  const int lm   = lane & 15;
  const int c0   = half * 2;

  int gtile = blockIdx.x * 4 + wave;
  if (gtile >= GTILES_) return;
  const int b  = gtile / TILES_;
  const int r0 = (gtile % TILES_) * 16;
  const float* Mb = M + (size_t)b * N_ * K_;
  float* pb = pbuf[wave];

  v2f aM[4];
#pragma unroll
  for (int s = 0; s < 4; ++s) {
    const float* p = Mb + (size_t)(r0 + lm) * K_ + 4 * s + c0;
    aM[s].x = p[0];
    aM[s].y = p[1];
  }

  float rm[8], inv[8];
#pragma unroll
  for (int e = 0; e < 8; ++e) {
    rm[e]  = rmaxg[b * N_ + r0 + e + 8 * half];
    inv[e] = 1.0f / rsumg[b * N_ + r0 + e + 8 * half];
  }

  v8f num[2];
  num[0] = 0.0f;
  num[1] = 0.0f;

  for (int m0 = 0; m0 < N_; m0 += 16) {
    if (m0 + 16 < N_) __builtin_prefetch(Mb + (size_t)(m0 + 16) * K_, 0, 1);

    v2f bM[4];
#pragma unroll
    for (int s = 0; s < 4; ++s) {
      const float* p = Mb + (size_t)(m0 + lm) * K_ + 4 * s + c0;
      bM[s].x = p[0];
      bM[s].y = p[1];
    }
    v8f Lv = 0.0f;
#pragma unroll
    for (int s = 0; s < 4; ++s) Lv = wmma4(aM[s], bM[s], Lv);

    float p[8];
#pragma unroll
    for (int e = 0; e < 8; ++e) p[e] = __expf(fmaxf(Lv[e], 0.f) - rm[e]);

#pragma unroll
    for (int e = 0; e < 8; ++e) pb[(e + 8 * half) * 17 + lm] = p[e];
    v2f pa[4];
#pragma unroll
    for (int s = 0; s < 4; ++s) {
      pa[s].x = pb[lm * 17 + 4 * s + c0];
      pa[s].y = pb[lm * 17 + 4 * s + c0 + 1];
    }

#pragma unroll
    for (int t = 0; t < 2; ++t)
#pragma unroll
      for (int s = 0; s < 4; ++s) {
        const float* q =
            Y2 + ((size_t)(b * N_ + m0 + 4 * s + c0)) * FOUT_ + 16 * t + lm;
        v2f by;
        by.x = q[0];
        by.y = q[FOUT_];
        num[t] = wmma4(pa[s], by, num[t]);
      }
  }

#pragma unroll
  for (int t = 0; t < 2; ++t)
#pragma unroll
    for (int e = 0; e < 8; ++e) {
      int row = r0 + e + 8 * half, col = 16 * t + lm;
      size_t off = ((size_t)b * N_ + row) * FOUT_ + col;
      out[off] = Y2[off] - num[t][e] * inv[e] + b2[col];
    }
}

extern "C" void kernel_launch(void* const* d_in, const int* in_sizes, int n_in,
                              void* d_out, int out_size, void* d_ws,
                              size_t ws_size, hipStream_t stream) {
  const float* X  = (const float*)d_in[0];
  const float* M  = (const float*)d_in[1];
  const float* W1 = (const float*)d_in[2];
  const float* b1 = (const float*)d_in[3];
  const float* W2 = (const float*)d_in[4];
  const float* b2 = (const float*)d_in[5];
  float* out = (float*)d_out;

  float* ws = (float*)d_ws;
  float* Y1    = ws;                            // B*N*64
  float* Y2    = Y1 + (size_t)B_ * N_ * FHID_;  // B*N*32
  float* rmaxg = Y2 + (size_t)B_ * N_ * FOUT_;  // B*N
  float* rsumg = rmaxg + (size_t)B_ * N_;       // B*N

  {
    int total = B_ * N_ * FHID_;
    k_y1<<<(total + 255) / 256, 256, 0, stream>>>(X, W1, Y1);
  }
  {
    int blocks = (GTILES_ + 3) / 4;  // 4 waves per 128-thread block
    k_layer1<<<blocks, 128, 0, stream>>>(M, Y1, W2, b1, Y2, rmaxg, rsumg);
    k_layer2<<<blocks, 128, 0, stream>>>(M, Y2, b2, rmaxg, rsumg, out);
  }
}